// WassersteinLoss_35622458753906
// MI455X (gfx1250) — compile-verified
//
#include <hip/hip_runtime.h>

// Wasserstein-with-END-mask reduction for MI455X (gfx1250, wave32).
//
// Strategy: one 256-thread workgroup per row. Scan the row in 1024-float
// tiles; per tile, detect the first END token via wave32 ballot + one LDS
// exchange, accumulate masked |real-gen| (gen loaded only where the mask is
// non-zero), and early-exit once the END tile is processed. Expected HBM
// traffic ~36MB instead of 256MB (END prob = 1/50 -> prefix ~49 elements).
// Block-level 256-way sum is done exactly in f32 on the matrix pipe with
// V_WMMA_F32_16X16X4_F32 (B = ones => D column 0 = row sums of A).
// Final deterministic tree reduction of the 8192 block partials.

#define COLS    4096
#define TPB     256
#define NWAVES  (TPB / 32)        // 8 waves (wave32)
#define TILE    (TPB * 4)         // 1024 floats per tile
#define NTILES  (COLS / TILE)     // 4 tiles per row
#define END_TOK 2.0f

typedef __attribute__((ext_vector_type(2))) float v2f;
typedef __attribute__((ext_vector_type(8))) float v8f;

__global__ __launch_bounds__(TPB)
void wass_row_kernel(const float* __restrict__ real,
                     const float* __restrict__ gen,
                     float* __restrict__ partial) {
    __shared__ unsigned s_fl[NWAVES];   // per-wave first-END lane (32 = none)
    __shared__ float    s_vals[TPB];    // per-thread masked partial sums

    const int tid  = threadIdx.x;
    const int lane = tid & 31;
    const int wave = tid >> 5;
    const long long rowBase = (long long)blockIdx.x * COLS;

    float threadTotal = 0.0f;

    for (int t = 0; t < NTILES; ++t) {
        const long long idx = rowBase + (long long)t * TILE + tid * 4;
        const float4 r = *reinterpret_cast<const float4*>(real + idx);

        // Local first-END position within this thread's 4 elements (4 = none).
        int e = 4;
        if      (r.x == END_TOK) e = 0;
        else if (r.y == END_TOK) e = 1;
        else if (r.z == END_TOK) e = 2;
        else if (r.w == END_TOK) e = 3;

        // wave32 ballot -> first lane in this wave holding an END token.
        const unsigned bal = __builtin_amdgcn_ballot_w32(e < 4);
        const unsigned fl  = bal ? (unsigned)__builtin_ctz(bal) : 32u;
        if (lane == 0) s_fl[wave] = fl;
        __syncthreads();

        // First wave (in thread order) containing an END token.
        unsigned wstar = NWAVES;
        for (int w = 0; w < NWAVES; ++w) {
            if (s_fl[w] < 32u) { wstar = (unsigned)w; break; }
        }

        // Number of this thread's elements that are inside the valid prefix.
        // waves < wstar: no END in them => e == 4 (full). wave == wstar:
        // lanes < fl have e == 4; lane == fl contributes e; lanes > fl: 0.
        const int limit = (((unsigned)wave <  wstar) ||
                           ((unsigned)wave == wstar && (unsigned)lane <= fl)) ? e : 0;

        if (limit > 0) {
            // gen is loaded only where the mask is non-zero (saves ~half the
            // HBM traffic given the expected ~49-element valid prefix).
            const float4 g = *reinterpret_cast<const float4*>(gen + idx);
            float ls = fabsf(r.x - g.x);
            if (limit > 1) ls += fabsf(r.y - g.y);
            if (limit > 2) ls += fabsf(r.z - g.z);
            if (limit > 3) ls += fabsf(r.w - g.w);
            threadTotal += ls;
        }

        __syncthreads();                  // s_fl reused next tile
        if (wstar < NWAVES) break;        // uniform: END tile processed, done
    }

    s_vals[tid] = threadTotal;
    __syncthreads();

    // Block reduction of 256 f32 partials on the matrix pipe, exact in f32:
    // D = A x ones + C  =>  D[M][N] = sum_K A[M][K] + C[M][N].
    // 4 chunks of 64 values (A is 16x4 f32 = 2 VGPRs per lane), accumulate C;
    // column N=0 (lanes 0 and 16) then holds the 16 row-sums of everything.
    if (wave == 0) {
        v8f acc = {0.0f, 0.0f, 0.0f, 0.0f, 0.0f, 0.0f, 0.0f, 0.0f};
        const v2f ones = {1.0f, 1.0f};
        for (int c = 0; c < TPB / 64; ++c) {
            v2f a;
            a.x = s_vals[c * 64 + lane * 2];
            a.y = s_vals[c * 64 + lane * 2 + 1];
            acc = __builtin_amdgcn_wmma_f32_16x16x4_f32(
                /*neg_a=*/false, a, /*neg_b=*/false, ones,
                /*c_mod=*/(short)0, acc, /*reuse_a=*/false, /*reuse_b=*/false);
        }
        // Per-lane: sum the 8 D VGPRs; column N=0 lives in lane 0 (M=0..7)
        // and lane 16 (M=8..15).
        float s = acc[0] + acc[1] + acc[2] + acc[3] +
                  acc[4] + acc[5] + acc[6] + acc[7];
        const float s16 = __int_as_float(
            __builtin_amdgcn_ds_bpermute(16 << 2, __float_as_int(s)));
        if (lane == 0) partial[blockIdx.x] = s + s16;
    }
}

__global__ __launch_bounds__(256)
void wass_final_kernel(const float* __restrict__ partial,
                       float* __restrict__ out, int n) {
    __shared__ float s[256];
    const int tid = threadIdx.x;
    float acc = 0.0f;
    for (int i = tid; i < n; i += 256) acc += partial[i];   // fixed order
    s[tid] = acc;
    __syncthreads();
    for (int off = 128; off > 0; off >>= 1) {               // fixed tree
        if (tid < off) s[tid] += s[tid + off];
        __syncthreads();
    }
    if (tid == 0) out[0] = s[0];
}

extern "C" void kernel_launch(void* const* d_in, const int* in_sizes, int n_in,
                              void* d_out, int out_size, void* d_ws, size_t ws_size,
                              hipStream_t stream) {
    const float* real = (const float*)d_in[0];   // real_distribution [B, S] f32
    const float* gen  = (const float*)d_in[1];   // generated_distribution [B, S] f32
    float* ws  = (float*)d_ws;                   // one partial per row
    float* out = (float*)d_out;                  // scalar f32

    const int rows = in_sizes[0] / COLS;         // B = 8192

    wass_row_kernel<<<rows, TPB, 0, stream>>>(real, gen, ws);
    wass_final_kernel<<<1, 256, 0, stream>>>(ws, out, rows);
}